// GACRF_11742440587460
// MI455X (gfx1250) — compile-verified
//
#include <hip/hip_runtime.h>

// GACRF fused kernel for MI455X (gfx1250, wave32).
// out[b,c,y,x] = logit - E^T * CRF3x3(F, E * softmax_C(logit))
//
// Fused single pass: HBM traffic ~= 80MB logits + 37.75MB F + 80MB out
// (+L2-resident logit re-read) ~= 200MB -> ~8.6us floor @ 23.3 TB/s.
// Decode einsum (G=8 -> C=19 per pixel) runs on the matrix units via
// V_WMMA_F32_16X16X4_F32 (fp32-exact), everything else VALU + LDS.

#define B_  4
#define C_  19
#define G_  8
#define H_  512
#define W_  512
#define TW  32
#define TH  8
#define HW_T (TW + 2)        // 34 halo width
#define HH_T (TH + 2)        // 10 halo height
#define NPIX   (TW * TH)     // 256 output pixels / block
#define NHALO  (HW_T * HH_T) // 340 halo pixels / block
#define NTHREADS 256

#define QP    9              // group stride (pad 8->9, conflict-free)
#define CPE   32             // E column stride (zero-padded for WMMA B)
#define CPO   33             // sOut channel stride (conflict-free reads)

// LDS layout in floats
#define OFF_E   0
#define SZ_E    (G_ * CPE)       // 256
#define OFF_QG  (OFF_E + SZ_E)
#define SZ_QG   (NHALO * QP)     // 3060
#define OFF_RG  (OFF_QG + SZ_QG)
#define SZ_RG   (NPIX * QP)      // 2304
#define OFF_OUT (OFF_RG + SZ_RG)
#define SZ_OUT  (NPIX * CPO)     // 8448
#define SMEM_FLOATS (OFF_OUT + SZ_OUT)
#define SMEM_BYTES  (SMEM_FLOATS * 4)   // 56272 B

typedef __attribute__((ext_vector_type(2))) float v2f;
typedef __attribute__((ext_vector_type(8))) float v8f;

extern "C" __global__ __launch_bounds__(NTHREADS)
void gacrf_fused(const float* __restrict__ Fw,
                 const float* __restrict__ logit,
                 const float* __restrict__ mat,
                 float* __restrict__ out)
{
    extern __shared__ float smem[];
    float* sE   = smem + OFF_E;    // [G_][CPE], zero padded columns 19..31
    float* sQg  = smem + OFF_QG;   // [NHALO][QP] encoded halo tile
    float* sRg  = smem + OFF_RG;   // [NPIX][QP]  CRF result per group
    float* sOut = smem + OFF_OUT;  // [NPIX][CPO] decoded channels

    const int tid = threadIdx.x;
    const int x0  = blockIdx.x * TW;
    const int y0  = blockIdx.y * TH;
    const int bb  = blockIdx.z;
    const size_t HWs = (size_t)H_ * W_;

    // ---- Phase 0a: zero-pad E storage (so WMMA B fragments are branch-free)
    sE[tid] = 0.0f;   // SZ_E == NTHREADS == 256
    __syncthreads();

    // ---- Phase 0b: E[g,c] = softmax_G(100*matrix)[g,c]; one column per thread
    if (tid < C_) {
        const int c = tid;
        float m[G_], mx = -3.4e38f;
        #pragma unroll
        for (int g = 0; g < G_; ++g) { m[g] = 100.0f * mat[g * C_ + c]; mx = fmaxf(mx, m[g]); }
        float s = 0.0f;
        #pragma unroll
        for (int g = 0; g < G_; ++g) { m[g] = __expf(m[g] - mx); s += m[g]; }
        const float r = 1.0f / s;
        #pragma unroll
        for (int g = 0; g < G_; ++g) sE[g * CPE + c] = m[g] * r;
    }
    __syncthreads();

    // ---- Phase A: per-pixel softmax over C + encode C->G, over halo tile
    for (int i = tid; i < NHALO; i += NTHREADS) {
        const int hy = i / HW_T, hx = i % HW_T;
        const int y = y0 - 1 + hy, x = x0 - 1 + hx;
        float qg[G_];
        #pragma unroll
        for (int g = 0; g < G_; ++g) qg[g] = 0.0f;
        if (y >= 0 && y < H_ && x >= 0 && x < W_) {
            const float* lp = logit + ((size_t)bb * C_) * HWs + (size_t)y * W_ + x;
            float q[C_], mx = -3.4e38f;
            #pragma unroll
            for (int c = 0; c < C_; ++c) { q[c] = lp[(size_t)c * HWs]; mx = fmaxf(mx, q[c]); }
            float s = 0.0f;
            #pragma unroll
            for (int c = 0; c < C_; ++c) { q[c] = __expf(q[c] - mx); s += q[c]; }
            const float r = 1.0f / s;
            #pragma unroll
            for (int c = 0; c < C_; ++c) {
                const float qc = q[c] * r;          // Q[c] at this pixel
                #pragma unroll
                for (int g = 0; g < G_; ++g) qg[g] = fmaf(sE[g * CPE + c], qc, qg[g]);
            }
        }
        #pragma unroll
        for (int g = 0; g < G_; ++g) sQg[i * QP + g] = qg[g];
    }
    __syncthreads();

    // ---- Phase B: per-pixel 3x3 dynamic filter (weights F) over groups
    for (int p = tid; p < NPIX; p += NTHREADS) {
        const int ty = p / TW, tx = p % TW;
        const int y = y0 + ty, x = x0 + tx;
        const float* fp = Fw + ((size_t)bb * 9) * HWs + (size_t)y * W_ + x;
        float f[9];
        #pragma unroll
        for (int k = 0; k < 9; ++k) f[k] = fp[(size_t)k * HWs];
        float rg[G_];
        #pragma unroll
        for (int g = 0; g < G_; ++g) rg[g] = 0.0f;
        #pragma unroll
        for (int di = 0; di < 3; ++di)
            #pragma unroll
            for (int dj = 0; dj < 3; ++dj) {
                const int base = ((ty + di) * HW_T + (tx + dj)) * QP;
                const float fk = f[di * 3 + dj];
                #pragma unroll
                for (int g = 0; g < G_; ++g) rg[g] = fmaf(fk, sQg[base + g], rg[g]);
            }
        #pragma unroll
        for (int g = 0; g < G_; ++g) sRg[p * QP + g] = rg[g];
    }
    __syncthreads();

    // ---- Phase C: decode G->C with V_WMMA_F32_16X16X4_F32.
    // D[16 pix x 16 chan] += A[16 pix x 4 g] * B[4 g x 16 chan]
    // fp32 A layout (ISA 7.12.2): elem v of lane L -> A[M=L&15, K=v+(L>>4)*2]
    // fp32 B layout (mirror):     elem v of lane L -> B[K=v+(L>>4)*2, N=L&15]
    // fp32 D layout:              VGPR r of lane L -> D[M=r+(L>>4)*8, N=L&15]
    {
        const int wave  = tid >> 5;
        const int lane  = tid & 31;
        const int mrow  = lane & 15;
        const int khalf = lane >> 4;
        for (int mt = wave; mt < NPIX / 16; mt += NTHREADS / 32) {
            const int pixbase = mt * 16;
            const int arow = (pixbase + mrow) * QP;
            v2f a0, a1;                         // K = g: 0..3 then 4..7
            a0.x = sRg[arow + 0 + khalf * 2];
            a0.y = sRg[arow + 1 + khalf * 2];
            a1.x = sRg[arow + 4 + khalf * 2];
            a1.y = sRg[arow + 5 + khalf * 2];
            #pragma unroll
            for (int cb = 0; cb < 2; ++cb) {    // N chunks: channels 0..15, 16..31(pad)
                const int cc = cb * 16 + mrow;  // this lane's channel column (< CPE)
                v2f b0, b1;
                b0.x = sE[(0 + khalf * 2) * CPE + cc];
                b0.y = sE[(1 + khalf * 2) * CPE + cc];
                b1.x = sE[(4 + khalf * 2) * CPE + cc];
                b1.y = sE[(5 + khalf * 2) * CPE + cc];
                v8f acc = {};
                acc = __builtin_amdgcn_wmma_f32_16x16x4_f32(
                          false, a0, false, b0, (short)0, acc, false, false);
                acc = __builtin_amdgcn_wmma_f32_16x16x4_f32(
                          false, a1, false, b1, (short)0, acc, false, false);
                // Unconditional store: columns >= C_ land in sOut pad (CPO=33)
                #pragma unroll
                for (int r = 0; r < 8; ++r) {
                    const int pixel = pixbase + r + khalf * 8;
                    sOut[pixel * CPO + cc] = acc[r];
                }
            }
        }
    }
    __syncthreads();

    // ---- Phase D: out = logit - decoded; coalesced global RW, stride-33 LDS
    for (int idx = tid; idx < NPIX * C_; idx += NTHREADS) {
        const int c = idx >> 8;          // NPIX == 256
        const int p = idx & 255;
        const int ty = p / TW, tx = p % TW;
        const size_t gi = ((size_t)bb * C_ + c) * HWs + (size_t)(y0 + ty) * W_ + (x0 + tx);
        out[gi] = logit[gi] - sOut[p * CPO + c];
    }
}

extern "C" void kernel_launch(void* const* d_in, const int* in_sizes, int n_in,
                              void* d_out, int out_size, void* d_ws, size_t ws_size,
                              hipStream_t stream) {
    (void)in_sizes; (void)n_in; (void)d_ws; (void)ws_size; (void)out_size;
    const float* dF = (const float*)d_in[0];   // [4,9,512,512]
    const float* dL = (const float*)d_in[1];   // [4,19,512,512]
    const float* dM = (const float*)d_in[2];   // [8,19,1,1]
    float* dO = (float*)d_out;                 // [4,19,512,512]
    dim3 grid(W_ / TW, H_ / TH, B_);           // 16 x 64 x 4 = 4096 blocks
    gacrf_fused<<<grid, NTHREADS, SMEM_BYTES, stream>>>(dF, dL, dM, dO);
}